// Fusin_Dice_rank_7095285973219
// MI455X (gfx1250) — compile-verified
//
#include <hip/hip_runtime.h>
#include <hip/hip_bf16.h>
#include <stdint.h>

// ---------------- problem constants ----------------
#define B_SAMPLES 32
#define HW        262144            // 512*512
#define NBLK      32                // streaming blocks per sample
#define CHUNK     (HW / NBLK)       // 8192 elements per block
#define ITERS     (CHUNK / 1024)    // 8 pipeline iterations (256 thr * 4 f32)
#define NTHR      256
#define TOPK      30
#define NSHELL    16
#define CAP       1024              // candidate buffer per sample per list
#define RPT       (CAP / NTHR)      // 4 candidate regs per thread
#define SMOOTH_F  1e-5f
#define MARGIN_F  0.3f

#define AS1 __attribute__((address_space(1)))
#define AS3 __attribute__((address_space(3)))

// ---------------- CDNA5 feature probes ----------------
#if defined(__has_builtin)
#  if __has_builtin(__builtin_amdgcn_global_load_async_to_lds_b128) && \
      __has_builtin(__builtin_amdgcn_s_wait_asynccnt)
#    define HAVE_ASYNC 1
#  endif
#  if __has_builtin(__builtin_amdgcn_wmma_f32_16x16x4_f32)
#    define HAVE_WMMA4 1
#  endif
#  if __has_builtin(__builtin_amdgcn_rcpf)
#    define HAVE_RCP 1
#  endif
#endif
#ifndef HAVE_ASYNC
#define HAVE_ASYNC 0
#endif
#ifndef HAVE_WMMA4
#define HAVE_WMMA4 0
#endif
#ifndef HAVE_RCP
#define HAVE_RCP 0
#endif

typedef __attribute__((ext_vector_type(8))) float v8f;
typedef __attribute__((ext_vector_type(2))) float v2f;
// Builtin's parameter type is a GCC-style 16B int vector (per diagnostic:
// "__attribute__((__vector_size__(4 * sizeof(int)))) int __device__ *").
typedef int gv4i __attribute__((vector_size(16)));

// 16B-per-lane async global->LDS copy (one instruction per wave -> ASYNCcnt+1).
// Generic shared ptr low 32 bits == wave-relative LDS address (ISA 10.2), so
// the AS3 pointer is built via integer truncation (avoids addrspacecast).
__device__ __forceinline__ void async_copy16(const float* g, float* l) {
#if HAVE_ASYNC
  __builtin_amdgcn_global_load_async_to_lds_b128(
      (AS1 gv4i*)(uintptr_t)g,
      (AS3 gv4i*)(unsigned)(uintptr_t)l, 0, 0);
#else
  *(float4*)l = *(const float4*)g;
#endif
}

#if HAVE_ASYNC
#define WAITA(n) __builtin_amdgcn_s_wait_asynccnt(n)
#else
#define WAITA(n) ((void)0)
#endif

// ---------------- WMMA-backed f32 sum accumulator ----------------
// D = A(16x4) * ones(4x16) + C : every column of C accumulates the full sum of
// the 64 packed A values, so (sum of all 8 C regs over all lanes)/16 == total.
struct Acc {
#if HAVE_WMMA4
  v8f c;
  __device__ __forceinline__ void init() { c = v8f{0,0,0,0,0,0,0,0}; }
  __device__ __forceinline__ void add2(float a, float b) {
    v2f av; av.x = a; av.y = b;
    v2f on; on.x = 1.0f; on.y = 1.0f;
    c = __builtin_amdgcn_wmma_f32_16x16x4_f32(false, av, false, on,
                                              (short)0, c, false, false);
  }
  __device__ __forceinline__ float total() const {
    float s = 0.0f;
#pragma unroll
    for (int i = 0; i < 8; ++i) s += c[i];
    return s * 0.0625f;   // /16: 16 identical columns
  }
#else
  float s_;
  __device__ __forceinline__ void init() { s_ = 0.0f; }
  __device__ __forceinline__ void add2(float a, float b) { s_ += a + b; }
  __device__ __forceinline__ float total() const { return s_; }
#endif
};

// ---------------- helpers ----------------
__device__ __forceinline__ float sigp1(float x0, float x1) {
  // softmax over 2 channels: p1 = 1/(1+exp(x0-x1)); saturates correctly.
  float den = 1.0f + __expf(x0 - x1);
#if HAVE_RCP
  return __builtin_amdgcn_rcpf(den);   // v_rcp_f32 (1 ulp), avoids div expansion
#else
  return 1.0f / den;
#endif
}

// shell j  <=>  v in (1-2^-j, 1-2^-(j+1)]   (v in [0,1))
__device__ __forceinline__ int shellbin(float v) {
  float u = 1.0f - v;
  int e = (int)((__float_as_uint(u) >> 23) & 0xFFu);
  int j = 126 - e;
  return j < 0 ? 0 : (j > 15 ? 15 : j);
}

__device__ __forceinline__ float block_reduce_sum(float v, float* s_red) {
  const int tid = threadIdx.x;
  s_red[tid] = v; __syncthreads();
  for (int s = NTHR / 2; s > 0; s >>= 1) {
    if (tid < s) s_red[tid] += s_red[tid + s];
    __syncthreads();
  }
  float r = s_red[0]; __syncthreads();
  return r;
}

// ---------------- kernel 0: zero the counters ----------------
__global__ void k_init(unsigned* z) {
  const int n = B_SAMPLES * NSHELL * 2 + B_SAMPLES * 2;   // 1088 uints
  for (int i = threadIdx.x; i < n; i += blockDim.x) z[i] = 0u;
}

// ---------------- kernel 1: main 96MB pass ----------------
__device__ __forceinline__ void hist_elem(float p1, float t,
                                          unsigned* hN, unsigned* hP) {
  float N = p1 * (1.0f - t);
  float P = (1.0f - p1) * t;
  if (N > 0.0f) atomicAdd(&hN[shellbin(N)], 1u);
  if (P > 0.0f) atomicAdd(&hP[shellbin(P)], 1u);
}

__device__ __forceinline__ void consume_iter(const float* sx0, const float* sx1,
                                             const float* st, Acc& aP, Acc& aT,
                                             Acc& aI, unsigned* hN, unsigned* hP) {
  float4 a = *(const float4*)sx0;
  float4 c = *(const float4*)sx1;
  float4 t = *(const float4*)st;
  float p0 = sigp1(a.x, c.x), p1 = sigp1(a.y, c.y);
  float p2 = sigp1(a.z, c.z), p3 = sigp1(a.w, c.w);
  hist_elem(p0, t.x, hN, hP); hist_elem(p1, t.y, hN, hP);
  hist_elem(p2, t.z, hN, hP); hist_elem(p3, t.w, hN, hP);
  aP.add2(p0, p1);            aP.add2(p2, p3);
  aT.add2(t.x, t.y);          aT.add2(t.z, t.w);
  aI.add2(p0 * t.x, p1 * t.y); aI.add2(p2 * t.z, p3 * t.w);
}

__global__ __launch_bounds__(NTHR) void k_main(const float* __restrict__ pred,
                                               const float* __restrict__ tgt,
                                               unsigned* __restrict__ histN,
                                               unsigned* __restrict__ histP,
                                               float* __restrict__ partials) {
  __shared__ __align__(16) float s_x0[2][1024];
  __shared__ __align__(16) float s_x1[2][1024];
  __shared__ __align__(16) float s_t [2][1024];
  __shared__ unsigned s_hN[NSHELL], s_hP[NSHELL];
  __shared__ float s_red[NTHR];

  const int tid = threadIdx.x;
  const int b   = blockIdx.x / NBLK;
  const int blk = blockIdx.x % NBLK;
  if (tid < NSHELL) { s_hN[tid] = 0u; s_hP[tid] = 0u; }
  __syncthreads();

  const float* x0 = pred + (size_t)b * (2 * HW) + (size_t)blk * CHUNK;
  const float* x1 = x0 + HW;
  const float* tt = tgt + (size_t)b * HW + (size_t)blk * CHUNK;

  Acc accP, accT, accI; accP.init(); accT.init(); accI.init();

#define K1_ISSUE(it) do {                                            \
    int _o = (it) * 1024 + tid * 4;                                  \
    async_copy16(x0 + _o, &s_x0[(it) & 1][tid * 4]);                 \
    async_copy16(x1 + _o, &s_x1[(it) & 1][tid * 4]);                 \
    async_copy16(tt + _o, &s_t [(it) & 1][tid * 4]); } while (0)

#define K1_STEP(it) do {                                             \
    if ((it) + 1 < ITERS) K1_ISSUE((it) + 1);                        \
    WAITA(((it) + 1 < ITERS) ? 3 : 0);                               \
    consume_iter(&s_x0[(it) & 1][tid * 4], &s_x1[(it) & 1][tid * 4], \
                 &s_t[(it) & 1][tid * 4], accP, accT, accI,          \
                 s_hN, s_hP); } while (0)

  K1_ISSUE(0);
  K1_STEP(0); K1_STEP(1); K1_STEP(2); K1_STEP(3);
  K1_STEP(4); K1_STEP(5); K1_STEP(6); K1_STEP(7);
#undef K1_STEP
#undef K1_ISSUE

  __syncthreads();
  if (tid < NSHELL) {
    atomicAdd(&histN[b * NSHELL + tid], s_hN[tid]);
    atomicAdd(&histP[b * NSHELL + tid], s_hP[tid]);
  }
  float rp = block_reduce_sum(accP.total(), s_red);
  float rt = block_reduce_sum(accT.total(), s_red);
  float ri = block_reduce_sum(accI.total(), s_red);
  if (tid == 0) {
    float* o = partials + ((size_t)b * NBLK + blk) * 3;
    o[0] = rp; o[1] = rt; o[2] = ri;
  }
}

// ---------------- kernel 2: candidate collection (L2-resident re-read) -------
__device__ __forceinline__ void collect_elem(float x0, float x1, float t,
                                             int jN, int jP,
                                             unsigned* cN, unsigned* cP,
                                             float* bN, float* bP) {
  float p = sigp1(x0, x1);
  float N = p * (1.0f - t);
  float P = (1.0f - p) * t;
  if (N > 0.0f && shellbin(N) >= jN) {
    unsigned i = atomicAdd(cN, 1u);
    if (i < CAP) bN[i] = N;
  }
  if (P > 0.0f && shellbin(P) >= jP) {
    unsigned i = atomicAdd(cP, 1u);
    if (i < CAP) bP[i] = P;
  }
}

__global__ __launch_bounds__(NTHR) void k_collect(const float* __restrict__ pred,
                                                  const float* __restrict__ tgt,
                                                  const unsigned* __restrict__ histN,
                                                  const unsigned* __restrict__ histP,
                                                  unsigned* __restrict__ cntN,
                                                  unsigned* __restrict__ cntP,
                                                  float* __restrict__ candN,
                                                  float* __restrict__ candP) {
  __shared__ __align__(16) float s_x0[2][1024];
  __shared__ __align__(16) float s_x1[2][1024];
  __shared__ __align__(16) float s_t [2][1024];
  __shared__ int s_j[2];

  const int tid = threadIdx.x;
  const int b   = blockIdx.x / NBLK;
  const int blk = blockIdx.x % NBLK;

  if (tid < 2) {   // smallest shell whose suffix count >= TOPK (else 0 = all >0)
    const unsigned* h = (tid == 0 ? histN : histP) + b * NSHELL;
    int suf = 0, j = 0;
    for (int i = NSHELL - 1; i >= 0; --i) {
      suf += (int)h[i];
      if (suf >= TOPK) { j = i; break; }
    }
    s_j[tid] = j;
  }
  __syncthreads();
  const int jN = s_j[0], jP = s_j[1];

  const float* x0 = pred + (size_t)b * (2 * HW) + (size_t)blk * CHUNK;
  const float* x1 = x0 + HW;
  const float* tt = tgt + (size_t)b * HW + (size_t)blk * CHUNK;
  unsigned* cN = cntN + b;
  unsigned* cP = cntP + b;
  float* bN = candN + (size_t)b * CAP;
  float* bP = candP + (size_t)b * CAP;

#define K3_ISSUE(it) do {                                            \
    int _o = (it) * 1024 + tid * 4;                                  \
    async_copy16(x0 + _o, &s_x0[(it) & 1][tid * 4]);                 \
    async_copy16(x1 + _o, &s_x1[(it) & 1][tid * 4]);                 \
    async_copy16(tt + _o, &s_t [(it) & 1][tid * 4]); } while (0)

#define K3_STEP(it) do {                                             \
    if ((it) + 1 < ITERS) K3_ISSUE((it) + 1);                        \
    WAITA(((it) + 1 < ITERS) ? 3 : 0);                               \
    {                                                                \
      float4 a = *(const float4*)&s_x0[(it) & 1][tid * 4];           \
      float4 c = *(const float4*)&s_x1[(it) & 1][tid * 4];           \
      float4 t = *(const float4*)&s_t [(it) & 1][tid * 4];           \
      collect_elem(a.x, c.x, t.x, jN, jP, cN, cP, bN, bP);           \
      collect_elem(a.y, c.y, t.y, jN, jP, cN, cP, bN, bP);           \
      collect_elem(a.z, c.z, t.z, jN, jP, cN, cP, bN, bP);           \
      collect_elem(a.w, c.w, t.w, jN, jP, cN, cP, bN, bP);           \
    } } while (0)

  K3_ISSUE(0);
  K3_STEP(0); K3_STEP(1); K3_STEP(2); K3_STEP(3);
  K3_STEP(4); K3_STEP(5); K3_STEP(6); K3_STEP(7);
#undef K3_STEP
#undef K3_ISSUE
}

// ---------------- kernel 3: per-sample exact top-30 + dice + rank ------------
__device__ void select_top(const float* cand, unsigned m, float* s_top,
                           float* s_val, int* s_idx, int* s_win) {
  const int tid = threadIdx.x;
  float v[RPT];
#pragma unroll
  for (int r = 0; r < RPT; ++r) {
    unsigned i = (unsigned)tid + (unsigned)r * NTHR;
    v[r] = (i < m) ? cand[i] : 0.0f;   // 0 sentinel matches top_k over >=0 data
  }
  unsigned used = 0;
  for (int round = 0; round < TOPK; ++round) {
    float bv = -1.0f; int bi = 0;
#pragma unroll
    for (int r = 0; r < RPT; ++r) {
      float x = (used & (1u << r)) ? -1.0f : v[r];
      if (x > bv) { bv = x; bi = r; }
    }
    s_val[tid] = bv; s_idx[tid] = tid;
    __syncthreads();
    for (int s = NTHR / 2; s > 0; s >>= 1) {
      if (tid < s && s_val[tid + s] > s_val[tid]) {
        s_val[tid] = s_val[tid + s]; s_idx[tid] = s_idx[tid + s];
      }
      __syncthreads();
    }
    if (tid == 0) { s_top[round] = fmaxf(s_val[0], 0.0f); *s_win = s_idx[0]; }
    __syncthreads();
    if (tid == *s_win && bv >= 0.0f) used |= (1u << bi);
    __syncthreads();
  }
}

__global__ __launch_bounds__(NTHR) void k_sample(const unsigned* __restrict__ cntN,
                                                 const unsigned* __restrict__ cntP,
                                                 const float* __restrict__ candN,
                                                 const float* __restrict__ candP,
                                                 const float* __restrict__ partials,
                                                 float* __restrict__ sampleRes) {
  const int b = blockIdx.x;
  const int tid = threadIdx.x;
  __shared__ float s_val[NTHR];
  __shared__ int   s_idx[NTHR];
  __shared__ float s_topN[TOPK], s_topP[TOPK];
  __shared__ int   s_win;

  unsigned mN = cntN[b]; if (mN > CAP) mN = CAP;
  unsigned mP = cntP[b]; if (mP > CAP) mP = CAP;
  select_top(candN + (size_t)b * CAP, mN, s_topN, s_val, s_idx, &s_win);
  select_top(candP + (size_t)b * CAP, mP, s_topP, s_val, s_idx, &s_win);

  // dice (serial, deterministic order)
  if (tid == 0) {
    float sp = 0.0f, st = 0.0f, si = 0.0f;
    for (int k = 0; k < NBLK; ++k) {
      const float* p = partials + ((size_t)b * NBLK + k) * 3;
      sp += p[0]; st += p[1]; si += p[2];
    }
    float I1 = si;
    float U1 = sp + st + SMOOTH_F;
    float I0 = (float)HW - st - sp + si;                      // sum((1-p1)(1-t))
    float U0 = ((float)HW - sp) + ((float)HW - st) + SMOOTH_F;
    float dice = (1.0f - 2.0f * I0 / U0) + (1.0f - 2.0f * I1 / U1);
    sampleRes[b * 2 + 0] = dice;
  }

  // rank hinge: th[i,j] = N_top[i] - (1 - P_vals[j]) + margin
  float local = 0.0f;
  for (int p = tid; p < TOPK * TOPK; p += NTHR) {
    int i = p / TOPK, j = p % TOPK;
    float th = s_topN[i] - (1.0f - s_topP[j]) + MARGIN_F;
    local += fmaxf(th, 0.0f);
  }
  s_val[tid] = local; __syncthreads();
  for (int s = NTHR / 2; s > 0; s >>= 1) {
    if (tid < s) s_val[tid] += s_val[tid + s];
    __syncthreads();
  }
  if (tid == 0) sampleRes[b * 2 + 1] = s_val[0];
}

// ---------------- kernel 4: final scalars ----------------
__global__ void k_final(const float* __restrict__ sampleRes, float* __restrict__ out) {
  if (threadIdx.x == 0 && blockIdx.x == 0) {
    float d = 0.0f, r = 0.0f;
    for (int b = 0; b < B_SAMPLES; ++b) {
      d += sampleRes[2 * b + 0];
      r += sampleRes[2 * b + 1];
    }
    out[0] = d / (2.0f * (float)B_SAMPLES);
    out[1] = r / ((float)B_SAMPLES * (float)(TOPK * TOPK));
  }
}

// ---------------- host side ----------------
extern "C" void kernel_launch(void* const* d_in, const int* in_sizes, int n_in,
                              void* d_out, int out_size, void* d_ws, size_t ws_size,
                              hipStream_t stream) {
  (void)in_sizes; (void)n_in; (void)out_size; (void)ws_size;
  const float* pred = (const float*)d_in[0];   // [32,2,512,512] f32
  const float* tgt  = (const float*)d_in[1];   // [32,512,512]  f32
  float* out = (float*)d_out;                  // {loss_D, loss_R}

  // workspace carve (~273 KB total)
  uint8_t* w = (uint8_t*)d_ws;
  unsigned* histN   = (unsigned*)w; w += (size_t)B_SAMPLES * NSHELL * 4;
  unsigned* histP   = (unsigned*)w; w += (size_t)B_SAMPLES * NSHELL * 4;
  unsigned* cntN    = (unsigned*)w; w += (size_t)B_SAMPLES * 4;
  unsigned* cntP    = (unsigned*)w; w += (size_t)B_SAMPLES * 4;
  float*    partials  = (float*)w;  w += (size_t)B_SAMPLES * NBLK * 3 * 4;
  float*    sampleRes = (float*)w;  w += (size_t)B_SAMPLES * 2 * 4;
  float*    candN   = (float*)w;    w += (size_t)B_SAMPLES * CAP * 4;
  float*    candP   = (float*)w;    w += (size_t)B_SAMPLES * CAP * 4;

  k_init<<<1, NTHR, 0, stream>>>(histN);   // hist/cnt block is contiguous
  k_main<<<B_SAMPLES * NBLK, NTHR, 0, stream>>>(pred, tgt, histN, histP, partials);
  k_collect<<<B_SAMPLES * NBLK, NTHR, 0, stream>>>(pred, tgt, histN, histP,
                                                   cntN, cntP, candN, candP);
  k_sample<<<B_SAMPLES, NTHR, 0, stream>>>(cntN, cntP, candN, candP,
                                           partials, sampleRes);
  k_final<<<1, 32, 0, stream>>>(sampleRes, out);
}